// CriterionLSCos_13503377179168
// MI455X (gfx1250) — compile-verified
//
#include <hip/hip_runtime.h>

typedef __attribute__((ext_vector_type(2))) float v2f;
typedef __attribute__((ext_vector_type(8))) float v8f;

#define C_DIM   19
#define TILE_W  32
#define TILE_H  16
#define HALO_W  34
#define HALO_H  18
#define PITCH   36                        // padded row pitch (dwords), even
#define PLANE   (HALO_H * PITCH)          // 648
#define LDS_N   (C_DIM * PLANE)           // 12312 floats = 49248 B per tensor
#define NPAIR   (C_DIM * HALO_H * (HALO_W / 2))  // 5814 float2 per tensor
#define NBLOCKS 2048                      // 16 batches * 16 h-tiles * 8 w-tiles

// ---------------------------------------------------------------------------
// Async global -> LDS copies (gfx1250 GLOBAL_LOAD_ASYNC_TO_LDS_B64/B32).
// Builtin pointer types confirmed by clang diagnostics:
//   b32: (global int*, local int*, Ii offset, Ii cpol)
//   b64: (global int __vector(2)*, local int __vector(2)*, Ii, Ii)
// ISA: IOFFSET is added to BOTH global and LDS addresses, so the b32
// fallback reuses one address pair with offset 4.
// ---------------------------------------------------------------------------
typedef int v2i_vs __attribute__((vector_size(8)));
typedef __attribute__((address_space(1))) int*    gptr_t;
typedef __attribute__((address_space(3))) int*    lptr_t;
typedef __attribute__((address_space(1))) v2i_vs* g64_t;
typedef __attribute__((address_space(3))) v2i_vs* l64_t;

#if __has_builtin(__builtin_amdgcn_global_load_async_to_lds_b64)
#define HAVE_ASYNC_LDS 1
__device__ __forceinline__ void copy_pair(const float* g, float* l) {
  __builtin_amdgcn_global_load_async_to_lds_b64((g64_t)g, (l64_t)l, 0, 0);
}
#elif __has_builtin(__builtin_amdgcn_global_load_async_to_lds_b32)
#define HAVE_ASYNC_LDS 1
__device__ __forceinline__ void copy_pair(const float* g, float* l) {
  __builtin_amdgcn_global_load_async_to_lds_b32((gptr_t)g, (lptr_t)l, 0, 0);
  __builtin_amdgcn_global_load_async_to_lds_b32((gptr_t)g, (lptr_t)l, 4, 0);
}
#else
#define HAVE_ASYNC_LDS 0
__device__ __forceinline__ void copy_pair(const float* g, float* l) {
  l[0] = g[0];
  l[1] = g[1];
}
#endif

__device__ __forceinline__ void async_wait_all() {
#if HAVE_ASYNC_LDS
#if __has_builtin(__builtin_amdgcn_s_wait_asynccnt)
  __builtin_amdgcn_s_wait_asynccnt(0);
#else
  asm volatile("s_wait_asynccnt 0x0" ::: "memory");
#endif
#endif
}

// ---------------------------------------------------------------------------
// Exact-f32 wave32 sum using V_WMMA_F32_16X16X4_F32 (verified lowering).
// A(16x4): lane L<16 -> A[L][0]=v_L; lane L>=16 -> A[L-16][2]=v_L; rest 0.
// B(4x16) = ones  =>  D[m][n] = v_m + v_{m+16}.
// Lanes 0-15 hold rows 0..7 in their 8 acc VGPRs, lanes 16-31 rows 8..15;
// per-lane add tree + (lane0 + lane16) = exact wave sum.
// ---------------------------------------------------------------------------
__device__ __forceinline__ float wave_sum32(float v) {
#if __has_builtin(__builtin_amdgcn_wmma_f32_16x16x4_f32)
  v2f a; a[0] = v;    a[1] = 0.0f;
  v2f b; b[0] = 1.0f; b[1] = 1.0f;
  v8f c = {0.f, 0.f, 0.f, 0.f, 0.f, 0.f, 0.f, 0.f};
  v8f d = __builtin_amdgcn_wmma_f32_16x16x4_f32(
      /*neg_a=*/false, a, /*neg_b=*/false, b,
      /*c_mod=*/(short)0, c, /*reuse_a=*/false, /*reuse_b=*/false);
  float t = ((d[0] + d[1]) + (d[2] + d[3])) + ((d[4] + d[5]) + (d[6] + d[7]));
  return __shfl(t, 0, 32) + __shfl(t, 16, 32);
#else
  for (int off = 16; off > 0; off >>= 1) v += __shfl_down(v, off, 32);
  return __shfl(v, 0, 32);
#endif
}

// Compute the 8 neighbor-cosine values for one interior pixel from an LDS tile.
__device__ __forceinline__ void cos8(const float* __restrict__ t,
                                     int ly, int lx, float* __restrict__ out) {
  float cc = 0.f;
  float dot[8], nn[8];
#pragma unroll
  for (int o = 0; o < 8; ++o) { dot[o] = 0.f; nn[o] = 0.f; }

  const int doff[8] = { -PITCH - 1, -PITCH, -PITCH + 1,
                        -1,                  +1,
                        +PITCH - 1, +PITCH, +PITCH + 1 };

  for (int c = 0; c < C_DIM; ++c) {
    const float* p = t + c * PLANE + ly * PITCH + lx;
    const float ctr = p[0];
    cc = fmaf(ctr, ctr, cc);
#pragma unroll
    for (int o = 0; o < 8; ++o) {
      const float n = p[doff[o]];
      dot[o] = fmaf(ctr, n, dot[o]);
      nn[o]  = fmaf(n, n, nn[o]);
    }
  }
  const float rc = __builtin_amdgcn_rsqf(cc);
#pragma unroll
  for (int o = 0; o < 8; ++o)
    out[o] = dot[o] * rc * __builtin_amdgcn_rsqf(nn[o]);
}

// ---------------------------------------------------------------------------
// Main fused kernel: one 32x16 interior tile per block (2 pixels per thread),
// halo-tiled in LDS via async b64 copies, per-block partial sum to workspace.
// ---------------------------------------------------------------------------
__global__ __launch_bounds__(256) void cosmap_mse_partial(
    const float* __restrict__ preds, const float* __restrict__ soft,
    float* __restrict__ partial) {
  __shared__ float sP[LDS_N];
  __shared__ float sS[LDS_N];

  const int tx  = threadIdx.x;            // 0..31
  const int ty  = threadIdx.y;            // 0..7
  const int tid = ty * TILE_W + tx;
  const int blk = blockIdx.x;
  const int b   = blk >> 7;               // 128 tiles per batch
  const int th  = (blk >> 3) & 15;        // h-tile 0..15
  const int tw  = blk & 7;                // w-tile 0..7
  const int gh0 = th * TILE_H;            // halo origin (global coords)
  const int gw0 = tw * TILE_W;
  const size_t base = (size_t)b * (C_DIM * 256 * 256);

  // Cooperative halo fill, float2 granularity (coalesced along W; edge tiles
  // clamp to stay in-bounds — clamped values feed only guarded-off pixels).
  for (int i = tid; i < NPAIR; i += 256) {
    const int c  = i / (HALO_H * (HALO_W / 2));     // / 306
    const int r  = i - c * (HALO_H * (HALO_W / 2));
    const int y  = r / (HALO_W / 2);                // / 17
    const int x2 = (r - y * (HALO_W / 2)) * 2;      // 0,2,..,32
    int gh = gh0 + y;  if (gh > 255) gh = 255;
    int gw = gw0 + x2; if (gw > 254) gw = 254;      // keep the pair in-bounds
    const size_t off = base + ((size_t)c << 16) + (size_t)(gh << 8) + (size_t)gw;
    const int li = c * PLANE + y * PITCH + x2;      // 8B-aligned (all even)
    copy_pair(preds + off, &sP[li]);
    copy_pair(soft + off, &sS[li]);
  }
  async_wait_all();
  __syncthreads();

  float accum = 0.f;
  const int w = gw0 + tx + 1;             // interior coord, must be <= 254
#pragma unroll
  for (int rr = 0; rr < 2; ++rr) {
    const int ly = ty + 8 * rr + 1;
    float cP[8], cS[8];
    cos8(sP, ly, tx + 1, cP);
    cos8(sS, ly, tx + 1, cS);
    const int h = gh0 + ty + 8 * rr + 1;
    if (h <= 254 && w <= 254) {
#pragma unroll
      for (int o = 0; o < 8; ++o) {
        const float d = cP[o] - cS[o];
        accum = fmaf(d, d, accum);
      }
    }
  }

  // Wave-level exact f32 reduce via WMMA, then 8 wave sums -> block sum.
  const float wsum = wave_sum32(accum);
  __syncthreads();                        // LDS reads done; safe to reuse
  if ((tid & 31) == 0) sP[tid >> 5] = wsum;
  __syncthreads();
  if (tid == 0) {
    float s = 0.f;
#pragma unroll
    for (int i = 0; i < 8; ++i) s += sP[i];
    partial[blk] = s;
  }
}

// ---------------------------------------------------------------------------
// Deterministic final reduction: 2048 partials -> mean scalar.
// ---------------------------------------------------------------------------
__global__ __launch_bounds__(256) void cosmap_mse_final(
    const float* __restrict__ partial, float* __restrict__ out) {
  __shared__ float ws[8];
  const int tid = threadIdx.x;
  float s = 0.f;
  for (int i = tid; i < NBLOCKS; i += 256) s += partial[i];
  const float wsum = wave_sum32(s);
  if ((tid & 31) == 0) ws[tid >> 5] = wsum;
  __syncthreads();
  if (tid == 0) {
    float tot = 0.f;
#pragma unroll
    for (int i = 0; i < 8; ++i) tot += ws[i];
    *out = tot * (float)(1.0 / (16.0 * 8.0 * 254.0 * 254.0));
  }
}

extern "C" void kernel_launch(void* const* d_in, const int* in_sizes, int n_in,
                              void* d_out, int out_size, void* d_ws, size_t ws_size,
                              hipStream_t stream) {
  const float* preds = (const float*)d_in[0];
  const float* soft  = (const float*)d_in[1];
  float* partial = (float*)d_ws;          // NBLOCKS floats = 8 KB scratch
  (void)in_sizes; (void)n_in; (void)out_size; (void)ws_size;

  dim3 block(TILE_W, 8);
  cosmap_mse_partial<<<NBLOCKS, block, 0, stream>>>(preds, soft, partial);
  cosmap_mse_final<<<1, 256, 0, stream>>>(partial, (float*)d_out);
}